// DFineMultiscaleDeformableAttention_5652176961976
// MI455X (gfx1250) — compile-verified
//
#include <hip/hip_runtime.h>
#include <hip/hip_bf16.h>
#include <math.h>

// ---------------- problem constants (fixed by reference setup) ----------------
#define BATCH      16
#define NQ         300
#define D_MODEL    256
#define N_HEADS    8
#define HEAD_DIM   32
#define SUM_POINTS 16
#define SEQ        8500
#define NROWS      (BATCH * NQ)        // 4800
#define NCOLS      (2 * N_HEADS * SUM_POINTS + N_HEADS * SUM_POINTS) // 256 + 128 = 384
#define N_TILES_M  (NROWS / 16)        // 300
#define N_TILES_N  (NCOLS / 16)        // 24

typedef __attribute__((ext_vector_type(2))) float v2f;
typedef __attribute__((ext_vector_type(8))) float v8f;

// =============================================================================
// Kernel 1: fused projection GEMM  logits[4800 x 384] = hidden @ [w_off; w_attn]^T + bias
// One wave32 computes one 16x16 tile with V_WMMA_F32_16X16X4_F32 (exact fp32).
// =============================================================================
__global__ __launch_bounds__(256) void dfine_proj_wmma_kernel(
    const float* __restrict__ hidden,   // (4800, 256)
    const float* __restrict__ w_off,    // (256, 256)
    const float* __restrict__ b_off,    // (256)
    const float* __restrict__ w_attn,   // (128, 256)
    const float* __restrict__ b_attn,   // (128)
    float* __restrict__ logits)         // (4800, 384) workspace
{
    const int wave = (int)((blockIdx.x * blockDim.x + threadIdx.x) >> 5);
    const int lane = (int)(threadIdx.x & 31);
    const int tM = wave / N_TILES_N;
    const int tN = wave % N_TILES_N;
    if (tM >= N_TILES_M) return;

    const int lr = lane & 15;   // 0..15
    const int hi = lane >> 4;   // 0: K=[0,1], 1: K=[2,3] per WMMA f32 A/B layout

    const int arow = tM * 16 + lr;        // A-matrix row (query index) for this lane
    const int ncol = tN * 16 + lr;        // output feature column for this lane

    // out[m][n] = sum_k hidden[m][k] * W[n][k]  (W = w_off rows 0..255, w_attn rows 256..383)
    const float* wptr;
    const float* bptr;
    int wrow;
    if (tN < 16) { wptr = w_off;  bptr = b_off;  wrow = ncol; }
    else         { wptr = w_attn; bptr = b_attn; wrow = ncol - 256; }

    const float* aBase = hidden + (size_t)arow * D_MODEL + hi * 2;
    const float* bBase = wptr   + (size_t)wrow * D_MODEL + hi * 2;

    v8f acc = {};
#pragma unroll 8
    for (int k = 0; k < D_MODEL; k += 4) {
        v2f a = *(const v2f*)(aBase + k);
        v2f b = *(const v2f*)(bBase + k);
        // D = A(16x4) * B(4x16) + C ; fp32 exact
        acc = __builtin_amdgcn_wmma_f32_16x16x4_f32(
            /*neg_a=*/false, a, /*neg_b=*/false, b,
            /*c_mod=*/(short)0, acc, /*reuse_a=*/false, /*reuse_b=*/false);
    }

    const float bias = bptr[wrow];

    // C/D layout: lanes 0-15 -> M = r, lanes 16-31 -> M = r + 8 ; N = lane & 15
    const int mBase = tM * 16 + hi * 8;
#pragma unroll
    for (int r = 0; r < 8; ++r) {
        logits[(size_t)(mBase + r) * NCOLS + ncol] = acc[r] + bias;
    }
}

// =============================================================================
// Kernel 2: softmax + multiscale bilinear sampling.
// One wave32 per (b, q, head); lane = head channel (coalesced 128B corner reads).
// =============================================================================
__device__ __forceinline__ float corner_fetch(const float* __restrict__ encBH,
                                              int s0, int x, int y, int W, int H)
{
    // reference semantics: clip index, load, then mask by validity
    const bool valid = (x >= 0) & (x < W) & (y >= 0) & (y < H);
    const int xc = min(max(x, 0), W - 1);
    const int yc = min(max(y, 0), H - 1);
    const float v = encBH[(size_t)(s0 + yc * W + xc) * D_MODEL];
    return valid ? v : 0.0f;
}

__global__ __launch_bounds__(256) void dfine_sample_kernel(
    const float* __restrict__ enc,      // (16, 8500, 256)
    const float* __restrict__ refp,     // (16, 300, 1, 4)
    const float* __restrict__ logits,   // (4800, 384)
    float* __restrict__ out,            // (16, 300, 256)
    float* __restrict__ attn_out)       // (16, 300, 8, 16)
{
    const int wid  = (int)((blockIdx.x * blockDim.x + threadIdx.x) >> 5);
    const int lane = (int)(threadIdx.x & 31);
    if (wid >= BATCH * NQ * N_HEADS) return;

    const int head = wid % N_HEADS;
    const int bq   = wid / N_HEADS;     // 0..4799
    const int b    = bq / NQ;

    const float* lrow = logits + (size_t)bq * NCOLS;
    const float* arow = lrow + 2 * N_HEADS * SUM_POINTS + head * SUM_POINTS; // attn logits
    const float* orow = lrow + head * (SUM_POINTS * 2);                      // offsets (x,y)*16

    // ---- softmax over 16 points (wave-uniform, computed redundantly per lane) ----
    float lg[SUM_POINTS];
    float mx = -3.0e38f;
#pragma unroll
    for (int p = 0; p < SUM_POINTS; ++p) { lg[p] = arow[p]; mx = fmaxf(mx, lg[p]); }
    float ssum = 0.0f;
#pragma unroll
    for (int p = 0; p < SUM_POINTS; ++p) { lg[p] = __expf(lg[p] - mx); ssum += lg[p]; }
    const float inv = 1.0f / ssum;

    // ---- reference point (cx, cy, w, h) ----
    const float cx = refp[(size_t)bq * 4 + 0];
    const float cy = refp[(size_t)bq * 4 + 1];
    const float rw = refp[(size_t)bq * 4 + 2];
    const float rh = refp[(size_t)bq * 4 + 3];
    // offset scale = nps (1/4) * OFFSET_SCALE (0.5)
    const float sx = 0.25f * 0.5f * rw;
    const float sy = 0.25f * 0.5f * rh;

    const int   Hs[4] = {80, 40, 20, 10};
    const int   Ws[4] = {80, 40, 20, 10};
    const int   S0[4] = {0, 6400, 8000, 8400};

    const float* encBH = enc + (size_t)b * SEQ * D_MODEL + head * HEAD_DIM + lane;

    float acc = 0.0f;
#pragma unroll
    for (int p = 0; p < SUM_POINTS; ++p) {
        const int lvl = p >> 2;
        const int W = Ws[lvl], H = Hs[lvl], s0 = S0[lvl];

        const float w  = lg[p] * inv;
        const float lx = cx + orow[p * 2 + 0] * sx;   // sampling location in [0,1]
        const float ly = cy + orow[p * 2 + 1] * sy;
        // gx = ((2*loc - 1) + 1) * W/2 - 0.5 = loc*W - 0.5
        const float gx = lx * (float)W - 0.5f;
        const float gy = ly * (float)H - 0.5f;
        const float fx = floorf(gx);
        const float fy = floorf(gy);
        const float wx = gx - fx;
        const float wy = gy - fy;
        const int x0 = (int)fx;
        const int y0 = (int)fy;

        const float v00 = corner_fetch(encBH, s0, x0,     y0,     W, H);
        const float v01 = corner_fetch(encBH, s0, x0 + 1, y0,     W, H);
        const float v10 = corner_fetch(encBH, s0, x0,     y0 + 1, W, H);
        const float v11 = corner_fetch(encBH, s0, x0 + 1, y0 + 1, W, H);

        const float top = v00 + (v01 - v00) * wx;
        const float bot = v10 + (v11 - v10) * wx;
        acc += w * (top + (bot - top) * wy);
    }

    out[(size_t)bq * D_MODEL + head * HEAD_DIM + lane] = acc;

    // attn output: lanes 0..15 each recompute their own weight (no dynamic reg indexing)
    if (lane < SUM_POINTS) {
        attn_out[(size_t)bq * (N_HEADS * SUM_POINTS) + head * SUM_POINTS + lane] =
            __expf(arow[lane] - mx) * inv;
    }
}

// =============================================================================
// Host launcher
// =============================================================================
extern "C" void kernel_launch(void* const* d_in, const int* in_sizes, int n_in,
                              void* d_out, int out_size, void* d_ws, size_t ws_size,
                              hipStream_t stream) {
    const float* hidden = (const float*)d_in[0];   // (16, 300, 256)
    const float* enc    = (const float*)d_in[1];   // (16, 8500, 256)
    const float* refp   = (const float*)d_in[2];   // (16, 300, 1, 4)
    // d_in[3] = spatial_shapes (int64) -- fixed, hardcoded
    const float* w_off  = (const float*)d_in[4];   // (256, 256)
    const float* b_off  = (const float*)d_in[5];   // (256)
    const float* w_attn = (const float*)d_in[6];   // (128, 256)
    const float* b_attn = (const float*)d_in[7];   // (128)

    float* out_main = (float*)d_out;                       // (16,300,256)
    float* out_attn = out_main + (size_t)BATCH * NQ * D_MODEL; // (16,300,8,16)

    float* logits = (float*)d_ws;                          // (4800, 384) = 7.4 MB
    (void)ws_size; (void)in_sizes; (void)n_in; (void)out_size;

    // GEMM: 300 * 24 = 7200 tiles -> 7200 waves -> 900 blocks of 8 waves
    {
        const int waves  = N_TILES_M * N_TILES_N;
        const int blocks = (waves + 7) / 8;
        dfine_proj_wmma_kernel<<<blocks, 256, 0, stream>>>(
            hidden, w_off, b_off, w_attn, b_attn, logits);
    }

    // Sampling: 16*300*8 = 38400 waves -> 4800 blocks of 8 waves
    {
        const int waves  = BATCH * NQ * N_HEADS;
        const int blocks = (waves + 7) / 8;
        dfine_sample_kernel<<<blocks, 256, 0, stream>>>(
            enc, refp, logits, out_main, out_attn);
    }
}